// SimpleNN2_5506148074170
// MI455X (gfx1250) — compile-verified
//
#include <hip/hip_runtime.h>
#include <hip/hip_bf16.h>

// CDNA5 / gfx1250. wave32.
typedef float v2f __attribute__((ext_vector_type(2)));
typedef float v4f __attribute__((ext_vector_type(4)));
typedef float v8f __attribute__((ext_vector_type(8)));

// Read lane 0's value (wave-uniform broadcast of a WMMA D column-0 element).
static __device__ __forceinline__ float rl0(float v) {
    return __uint_as_float(__builtin_amdgcn_readlane(__float_as_uint(v), 0u));
}

// A-matrix (16x4 f32) VGPR layout per ISA 7.12.2:
//   lane L in [0,16):  v0 = A[L][cbase+0], v1 = A[L][cbase+1]
//   lane 16+L:         v0 = A[L][cbase+2], v1 = A[L][cbase+3]
// Rows/cols beyond the real matrix are zero-padded.
static __device__ __forceinline__ v2f load_A_tile(const float* __restrict__ W,
                                                  int rows, int cols, int cbase) {
    const int lane = (int)(threadIdx.x & 31u);
    const int r = lane & 15;
    const int c = cbase + ((lane >> 4) << 1);
    v2f a;
    a.x = (r < rows && (c + 0) < cols) ? W[r * cols + c + 0] : 0.0f;
    a.y = (r < rows && (c + 1) < cols) ? W[r * cols + c + 1] : 0.0f;
    return a;
}

// B-matrix (4x16 f32): broadcast vector x into every column:
//   v0 = { lanes0-15: B[0][n]=k0, lanes16-31: B[2][n]=k2 }
//   v1 = { lanes0-15: B[1][n]=k1, lanes16-31: B[3][n]=k3 }
// (Same K-slot convention as the A tile, so the contraction is exact.)
static __device__ __forceinline__ v2f make_B_colbcast(float k0, float k1,
                                                      float k2, float k3) {
    const bool hi = (threadIdx.x & 31u) >= 16u;
    v2f b;
    b.x = hi ? k2 : k0;
    b.y = hi ? k3 : k1;
    return b;
}

static __device__ __forceinline__ v8f wmma_f32_k4(v2f A, v2f B, v8f C) {
    // D = A(16x4) x B(4x16) + C ; emits v_wmma_f32_16x16x4_f32
    return __builtin_amdgcn_wmma_f32_16x16x4_f32(
        /*neg_a=*/false, A, /*neg_b=*/false, B,
        /*c_mod=*/(short)0, C, /*reuse_a=*/false, /*reuse_b=*/false);
}

// One wave. Collapses the 5-layer affine chain into ws[0..6]=w_eff, ws[7]=b_eff.
// Right-to-left mat-vec chain via V_WMMA_F32_16X16X4_F32.
__global__ __launch_bounds__(32) void collapse_kernel(
    const float* __restrict__ W1, const float* __restrict__ b1,
    const float* __restrict__ W2, const float* __restrict__ b2,
    const float* __restrict__ W3, const float* __restrict__ b3,
    const float* __restrict__ W4, const float* __restrict__ b4,
    const float* __restrict__ W5, const float* __restrict__ b5,
    float* __restrict__ ws) {
    const v8f z = {};

    // v5 = W5 (2x1)
    const float v5_0 = W5[0], v5_1 = W5[1];

    // v4 = W4(3x2) @ v5
    v8f D4 = wmma_f32_k4(load_A_tile(W4, 3, 2, 0),
                         make_B_colbcast(v5_0, v5_1, 0.f, 0.f), z);
    const float v4_0 = rl0(D4[0]), v4_1 = rl0(D4[1]), v4_2 = rl0(D4[2]);

    // v3 = W3(4x3) @ v4
    v8f D3 = wmma_f32_k4(load_A_tile(W3, 4, 3, 0),
                         make_B_colbcast(v4_0, v4_1, v4_2, 0.f), z);
    const float v3_0 = rl0(D3[0]), v3_1 = rl0(D3[1]);
    const float v3_2 = rl0(D3[2]), v3_3 = rl0(D3[3]);

    // v2 = W2(5x4) @ v3
    v8f D2 = wmma_f32_k4(load_A_tile(W2, 5, 4, 0),
                         make_B_colbcast(v3_0, v3_1, v3_2, v3_3), z);
    float v2v[5];
#pragma unroll
    for (int j = 0; j < 5; ++j) v2v[j] = rl0(D2[j]);

    // w_eff = W1(7x5) @ v2 : K=5 split as 4+1, chained through the C operand.
    v8f C1 = wmma_f32_k4(load_A_tile(W1, 7, 5, 4),
                         make_B_colbcast(v2v[4], 0.f, 0.f, 0.f), z);
    v8f D1 = wmma_f32_k4(load_A_tile(W1, 7, 5, 0),
                         make_B_colbcast(v2v[0], v2v[1], v2v[2], v2v[3]), C1);

    // b_eff = b1.v2 + b2.v3 + b3.v4 + b4.v5 + b5
    float beff = b5[0];
    beff = fmaf(b4[0], v5_0, beff);  beff = fmaf(b4[1], v5_1, beff);
    beff = fmaf(b3[0], v4_0, beff);  beff = fmaf(b3[1], v4_1, beff);
    beff = fmaf(b3[2], v4_2, beff);
    beff = fmaf(b2[0], v3_0, beff);  beff = fmaf(b2[1], v3_1, beff);
    beff = fmaf(b2[2], v3_2, beff);  beff = fmaf(b2[3], v3_3, beff);
#pragma unroll
    for (int j = 0; j < 5; ++j) beff = fmaf(b1[j], v2v[j], beff);

    // D column 0 lives in lane 0 (elements = w_eff rows 0..7; rows 7..15 are 0).
    float weff[7];
#pragma unroll
    for (int j = 0; j < 7; ++j) weff[j] = rl0(D1[j]);
    if ((threadIdx.x & 31u) == 0u) {
#pragma unroll
        for (int j = 0; j < 7; ++j) ws[j] = weff[j];
        ws[7] = beff;
    }
}

// Streaming kernel: each thread processes 4 rows = 28 floats = 7 x b128 loads
// (112 B, 16 B aligned), 7 FMA + sigmoid per row, one b128 store.
// Single-pass data -> non-temporal loads/stores (TH=NT) so the 192 MB L2 isn't
// churned. Memory bound: 128 MB traffic -> ~5.5 us floor at 23.3 TB/s.
__global__ __launch_bounds__(256) void mlp_stream_kernel(
    const float* __restrict__ X, const float* __restrict__ wb,
    float* __restrict__ out, long long rows) {
    const long long tid = (long long)blockIdx.x * blockDim.x + threadIdx.x;
    const long long nquad = rows >> 2;

    float w[8];
#pragma unroll
    for (int i = 0; i < 8; ++i) w[i] = wb[i];   // uniform -> scalar loads

    if (tid < nquad) {
        const v4f* xp = (const v4f*)(X + tid * 28ll);
        float f[28];
#pragma unroll
        for (int i = 0; i < 7; ++i) {
            v4f t = __builtin_nontemporal_load(xp + i);   // global_load_b128 TH_NT
            f[4 * i + 0] = t.x; f[4 * i + 1] = t.y;
            f[4 * i + 2] = t.z; f[4 * i + 3] = t.w;
        }
        v4f r;
#pragma unroll
        for (int j = 0; j < 4; ++j) {
            float h = w[7];
#pragma unroll
            for (int k = 0; k < 7; ++k) h = fmaf(f[7 * j + k], w[k], h);
            r[j] = 1.0f / (1.0f + __expf(-h));
        }
        __builtin_nontemporal_store(r, (v4f*)out + tid);  // global_store_b128 TH_NT
    } else if (tid == nquad) {
        // tail rows (rows % 4), scalar path
        for (long long r0 = nquad << 2; r0 < rows; ++r0) {
            float h = w[7];
#pragma unroll
            for (int k = 0; k < 7; ++k) h = fmaf(X[r0 * 7 + k], w[k], h);
            out[r0] = 1.0f / (1.0f + __expf(-h));
        }
    }
}

extern "C" void kernel_launch(void* const* d_in, const int* in_sizes, int n_in,
                              void* d_out, int out_size, void* d_ws, size_t ws_size,
                              hipStream_t stream) {
    const float* X  = (const float*)d_in[0];
    const float* W1 = (const float*)d_in[1];
    const float* b1 = (const float*)d_in[2];
    const float* W2 = (const float*)d_in[3];
    const float* b2 = (const float*)d_in[4];
    const float* W3 = (const float*)d_in[5];
    const float* b3 = (const float*)d_in[6];
    const float* W4 = (const float*)d_in[7];
    const float* b4 = (const float*)d_in[8];
    const float* W5 = (const float*)d_in[9];
    const float* b5 = (const float*)d_in[10];
    float* ws  = (float*)d_ws;
    float* out = (float*)d_out;

    const long long rows = (long long)in_sizes[0] / 7;   // 4,000,000

    collapse_kernel<<<1, 32, 0, stream>>>(W1, b1, W2, b2, W3, b3, W4, b4, W5, b5, ws);

    const long long nquad = rows >> 2;
    const long long nthreads = nquad + ((rows & 3) ? 1 : 0);
    const int blocks = (int)((nthreads + 255) / 256);
    mlp_stream_kernel<<<blocks, 256, 0, stream>>>(X, ws, out, rows);
}